// GridModel_71270687310572
// MI455X (gfx1250) — compile-verified
//
#include <hip/hip_runtime.h>

// Bilinear gather: out[i] = bilerp(weight, x[i], t[i]) on a 2048x2048 f32 grid.
// Memory-bound (AI ~ 1 flop/byte or less): optimized for the MI455X memory path.
//  - grid (16 MB) stays L2-resident (192 MB L2): grid taps use default RT hint
//  - x/t/out streams (192 MB total) use NT temporal hints so they do not
//    evict the grid from L2
//  - 4 points per thread -> b128 vector loads/stores for all streaming traffic
//  - wave32: 256-thread blocks = 8 waves/block, ~131K waves total for latency hiding

typedef __attribute__((ext_vector_type(4))) float v4f;

#define GRID_Y 2048
#define GRID_X 2048
#define LOG2_X 11

__device__ __forceinline__ float bilerp_one(const float* __restrict__ w,
                                            float xx, float tt) {
    // row index from x, col index from t (map_coordinates([x, t]) semantics)
    int r = (int)__builtin_floorf(xx);
    int c = (int)__builtin_floorf(tt);
    // clamp to dim-2: exact for all in-bounds coords incl. the x==2047.0 edge
    // (frac becomes 1.0 -> pure upper tap), and never reads OOB.
    r = r < 0 ? 0 : (r > GRID_Y - 2 ? GRID_Y - 2 : r);
    c = c < 0 ? 0 : (c > GRID_X - 2 ? GRID_X - 2 : c);
    float fx = xx - (float)r;
    float ft = tt - (float)c;
    const float* p = w + ((size_t)r << LOG2_X) + (size_t)c;
    // w00/w01 and w10/w11 are adjacent -> same cacheline, coalesce in L2.
    float w00 = p[0];
    float w01 = p[1];
    float w10 = p[GRID_X];
    float w11 = p[GRID_X + 1];
    float top = __builtin_fmaf(ft, w01 - w00, w00);
    float bot = __builtin_fmaf(ft, w11 - w10, w10);
    return __builtin_fmaf(fx, bot - top, top);
}

__global__ __launch_bounds__(256) void grid_bilerp_v4(
    const float* __restrict__ x,
    const float* __restrict__ t,
    const float* __restrict__ w,
    float* __restrict__ out,
    int n4) {
    int i = blockIdx.x * blockDim.x + threadIdx.x;
    if (i >= n4) return;

    // NT streaming loads: do not pollute L2 (keep the 16MB grid resident)
    v4f xv = __builtin_nontemporal_load((const v4f*)x + i);
    v4f tv = __builtin_nontemporal_load((const v4f*)t + i);

    v4f o;
#pragma unroll
    for (int k = 0; k < 4; ++k) {
        o[k] = bilerp_one(w, xv[k], tv[k]);
    }

    // NT streaming store
    __builtin_nontemporal_store(o, (v4f*)out + i);
}

// Scalar tail (only launched when N % 4 != 0; N = 2^24 in practice so unused,
// but keeps kernel_launch correct and deterministic for any size).
__global__ __launch_bounds__(64) void grid_bilerp_tail(
    const float* __restrict__ x,
    const float* __restrict__ t,
    const float* __restrict__ w,
    float* __restrict__ out,
    int start, int n) {
    int i = start + blockIdx.x * blockDim.x + threadIdx.x;
    if (i >= n) return;
    float r = bilerp_one(w, x[i], t[i]);
    __builtin_nontemporal_store(r, out + i);
}

extern "C" void kernel_launch(void* const* d_in, const int* in_sizes, int n_in,
                              void* d_out, int out_size, void* d_ws, size_t ws_size,
                              hipStream_t stream) {
    const float* x = (const float*)d_in[0];
    const float* t = (const float*)d_in[1];
    const float* w = (const float*)d_in[2];
    float* out = (float*)d_out;

    int n  = in_sizes[0];          // number of query points (2^24)
    int n4 = n >> 2;               // float4 chunks
    int rem = n & 3;

    if (n4 > 0) {
        int blocks = (n4 + 255) / 256;
        grid_bilerp_v4<<<blocks, 256, 0, stream>>>(x, t, w, out, n4);
    }
    if (rem > 0) {
        grid_bilerp_tail<<<1, 64, 0, stream>>>(x, t, w, out, n4 << 2, n);
    }
}